// HPGCN_66975720014344
// MI455X (gfx1250) — compile-verified
//
#include <hip/hip_runtime.h>
#include <hip/hip_bf16.h>
#include <math.h>

typedef __attribute__((ext_vector_type(2))) float v2f;
typedef __attribute__((ext_vector_type(8))) float v8f;

#define DIN 64
#define HID 64

__device__ __forceinline__ void atomAddF(float* p, float v) {
    __hip_atomic_fetch_add(p, v, __ATOMIC_RELAXED, __HIP_MEMORY_SCOPE_AGENT);
}

// ---------------- zero fill ----------------
__global__ void k_zero(float* __restrict__ p, int n) {
    int i = blockIdx.x * blockDim.x + threadIdx.x;
    if (i < n) p[i] = 0.0f;
}

// ---------------- degree counting: 4 arrays packed [src_pos|dst_pos|src_neg|dst_neg] ----------------
__global__ void k_count_deg(const int* __restrict__ sp, const int* __restrict__ dp,
                            const int* __restrict__ sn, const int* __restrict__ dn,
                            float* __restrict__ deg, int ne, int n) {
    int e = blockIdx.x * blockDim.x + threadIdx.x;
    if (e >= ne) return;
    atomAddF(&deg[sp[e]], 1.0f);
    atomAddF(&deg[n + dp[e]], 1.0f);
    atomAddF(&deg[2 * n + sn[e]], 1.0f);
    atomAddF(&deg[3 * n + dn[e]], 1.0f);
}

// ---------------- deg -> D^{-1/2} (zero-degree -> 0), in place ----------------
__global__ void k_norm(float* __restrict__ deg, int n4) {
    int i = blockIdx.x * blockDim.x + threadIdx.x;
    if (i >= n4) return;
    float d = deg[i];
    deg[i] = (d > 0.0f) ? rsqrtf(d) : 0.0f;
}

// ---------------- edge scatter: agg[dst, f] += h[src, f] * c_src[src] ----------------
// 64 threads per edge (feature dim); wave32 handles a contiguous half-row -> coalesced.
__global__ void k_scatter(const float* __restrict__ h, const float* __restrict__ csrc,
                          const int* __restrict__ src, const int* __restrict__ dst,
                          float* __restrict__ agg, int ne) {
    long long g = (long long)blockIdx.x * blockDim.x + threadIdx.x;
    int e = (int)(g >> 6);
    int f = (int)(g & 63);
    if (e >= ne) return;
    int s = src[e];
    int d = dst[e];
    float v = h[s * HID + f] * csrc[s];
    atomAddF(&agg[d * HID + f], v);
}

// ---------------- fused: out = ReLU( (agg * c_dst[:,None]) @ W + b ) via f32 WMMA ----------------
// One wave per 16-row stripe; 4 v8f accumulators cover all 64 output columns.
// K=64 consumed as 16 steps of V_WMMA_F32_16X16X4_F32.
// A-frag layout (ISA 7.12.2, 32-bit A 16x4): lanes 0-15 hold K={0,1}, lanes 16-31 hold K={2,3}.
// D layout (32-bit C/D 16x16): VGPR r -> M = r (lanes 0-15) / M = r+8 (lanes 16-31), N = lane&15.
__global__ void k_gemm_relu(const float* __restrict__ agg, const float* __restrict__ cdst,
                            const float* __restrict__ W, const float* __restrict__ bias,
                            float* __restrict__ out, int nrows) {
    int wave = (int)((blockIdx.x * blockDim.x + threadIdx.x) >> 5);
    int lane = threadIdx.x & 31;
    int m0 = wave * 16;
    if (m0 >= nrows) return;            // wave-uniform; EXEC stays all-1s for WMMA

    int half = lane >> 4;               // 0: lanes 0-15, 1: lanes 16-31
    int l = lane & 15;
    int arow = m0 + l;                  // A row served by this lane
    float cs = cdst[arow];
    int koff = half * 2;                // K sub-offset 0 or 2

    v8f acc0 = {}, acc1 = {}, acc2 = {}, acc3 = {};

    #pragma unroll
    for (int kk = 0; kk < HID; kk += 4) {
        v2f a;
        a.x = agg[arow * HID + kk + koff] * cs;
        a.y = agg[arow * HID + kk + koff + 1] * cs;

        const float* Wr0 = &W[(kk + koff) * HID];       // row K = kk+koff
        const float* Wr1 = &W[(kk + koff + 1) * HID];   // row K = kk+koff+1
        v2f b0, b1, b2, b3;
        b0.x = Wr0[l];       b0.y = Wr1[l];             // cols  0..15
        b1.x = Wr0[16 + l];  b1.y = Wr1[16 + l];        // cols 16..31
        b2.x = Wr0[32 + l];  b2.y = Wr1[32 + l];        // cols 32..47
        b3.x = Wr0[48 + l];  b3.y = Wr1[48 + l];        // cols 48..63

        acc0 = __builtin_amdgcn_wmma_f32_16x16x4_f32(false, a, false, b0, (short)0, acc0, false, false);
        acc1 = __builtin_amdgcn_wmma_f32_16x16x4_f32(false, a, false, b1, (short)0, acc1, false, false);
        acc2 = __builtin_amdgcn_wmma_f32_16x16x4_f32(false, a, false, b2, (short)0, acc2, false, false);
        acc3 = __builtin_amdgcn_wmma_f32_16x16x4_f32(false, a, false, b3, (short)0, acc3, false, false);
    }

    float bb0 = bias[l], bb1 = bias[16 + l], bb2 = bias[32 + l], bb3 = bias[48 + l];
    #pragma unroll
    for (int r = 0; r < 8; ++r) {
        int row = m0 + r + half * 8;
        float* o = &out[row * HID];
        o[l]      = fmaxf(acc0[r] + bb0, 0.0f);
        o[16 + l] = fmaxf(acc1[r] + bb1, 0.0f);
        o[32 + l] = fmaxf(acc2[r] + bb2, 0.0f);
        o[48 + l] = fmaxf(acc3[r] + bb3, 0.0f);
    }
}

// ---------------- z = z_pos - z_neg ----------------
__global__ void k_sub(const float* __restrict__ a, const float* __restrict__ b,
                      float* __restrict__ o, int n) {
    int i = blockIdx.x * blockDim.x + threadIdx.x;
    if (i < n) o[i] = a[i] - b[i];
}

// ---------------- edge classifier: sigmoid([z[s],z[d]] @ Wc + bc) ----------------
__global__ void k_classify(const float* __restrict__ z, const int* __restrict__ ei,
                           const float* __restrict__ Wc, const float* __restrict__ bc,
                           float* __restrict__ out, int eq) {
    int e = blockIdx.x * blockDim.x + threadIdx.x;
    if (e >= eq) return;
    int s = ei[e];
    int d = ei[eq + e];
    const float* zs = &z[s * HID];
    const float* zd = &z[d * HID];
    float l0 = bc[0], l1 = bc[1];
    #pragma unroll 8
    for (int k = 0; k < HID; ++k) {
        float vs = zs[k];
        float vd = zd[k];
        l0 += vs * Wc[k * 2 + 0] + vd * Wc[(HID + k) * 2 + 0];
        l1 += vs * Wc[k * 2 + 1] + vd * Wc[(HID + k) * 2 + 1];
    }
    out[e * 2 + 0] = 1.0f / (1.0f + __expf(-l0));
    out[e * 2 + 1] = 1.0f / (1.0f + __expf(-l1));
}

extern "C" void kernel_launch(void* const* d_in, const int* in_sizes, int n_in,
                              void* d_out, int out_size, void* d_ws, size_t ws_size,
                              hipStream_t stream) {
    // Inputs (setup_inputs order)
    const float* x      = (const float*)d_in[0];
    const float* W0_pos = (const float*)d_in[1];
    const float* b0_pos = (const float*)d_in[2];
    const float* W0_neg = (const float*)d_in[3];
    const float* b0_neg = (const float*)d_in[4];
    const float* W1_pos = (const float*)d_in[5];
    const float* b1_pos = (const float*)d_in[6];
    const float* W1_neg = (const float*)d_in[7];
    const float* b1_neg = (const float*)d_in[8];
    const float* Wc     = (const float*)d_in[9];
    const float* bc     = (const float*)d_in[10];
    const int* src_pos  = (const int*)d_in[11];
    const int* dst_pos  = (const int*)d_in[12];
    const int* src_neg  = (const int*)d_in[13];
    const int* dst_neg  = (const int*)d_in[14];
    const int* ei       = (const int*)d_in[15];

    const int N  = in_sizes[0] / DIN;       // 100000
    const int ER = in_sizes[11];            // 1.6M
    const int EQ = in_sizes[15] / 2;        // 1M
    const int NH = N * HID;

    // Workspace layout (floats)
    float* ws    = (float*)d_ws;
    float* norms = ws;                      // 4*N: [c_src_pos | c_dst_pos | c_src_neg | c_dst_neg]
    float* bufA  = norms + 4 * (size_t)N;   // N*64 aggregation scratch
    float* bufB  = bufA + (size_t)NH;       // N*64 pos stream (h_pos -> z_pos)
    float* bufC  = bufB + (size_t)NH;       // N*64 neg stream (h_neg -> z_neg)

    float* z_out   = (float*)d_out;         // first N*64 floats
    float* cls_out = z_out + (size_t)NH;    // next EQ*2 floats

    const int B = 256;
    dim3 blk(B);
    auto gridFor = [](long long n, int b) { return dim3((unsigned)((n + b - 1) / b)); };

    const int gemmWaves  = (N + 15) / 16;                 // one wave per 16 rows
    const dim3 gemmGrid  = gridFor((long long)gemmWaves * 32, B);
    const dim3 scatGrid  = gridFor((long long)ER * 64, B);

    // 1) degree norms
    k_zero<<<gridFor(4 * (long long)N, B), blk, 0, stream>>>(norms, 4 * N);
    k_count_deg<<<gridFor(ER, B), blk, 0, stream>>>(src_pos, dst_pos, src_neg, dst_neg, norms, ER, N);
    k_norm<<<gridFor(4 * (long long)N, B), blk, 0, stream>>>(norms, 4 * N);

    const float* c_sp = norms;
    const float* c_dp = norms + N;
    const float* c_sn = norms + 2 * N;
    const float* c_dn = norms + 3 * N;

    // 2) Layer 1, pos: h_pos = ReLU(conv(x) @ W0_pos + b0_pos) -> bufB
    k_zero<<<gridFor(NH, B), blk, 0, stream>>>(bufA, NH);
    k_scatter<<<scatGrid, blk, 0, stream>>>(x, c_sp, src_pos, dst_pos, bufA, ER);
    k_gemm_relu<<<gemmGrid, blk, 0, stream>>>(bufA, c_dp, W0_pos, b0_pos, bufB, N);

    // 3) Layer 1, neg -> bufC
    k_zero<<<gridFor(NH, B), blk, 0, stream>>>(bufA, NH);
    k_scatter<<<scatGrid, blk, 0, stream>>>(x, c_sn, src_neg, dst_neg, bufA, ER);
    k_gemm_relu<<<gemmGrid, blk, 0, stream>>>(bufA, c_dn, W0_neg, b0_neg, bufC, N);

    // 4) Layer 2, pos: z_pos = ReLU(conv(h_pos) @ W1_pos + b1_pos) -> bufB (scatter consumes bufB first)
    k_zero<<<gridFor(NH, B), blk, 0, stream>>>(bufA, NH);
    k_scatter<<<scatGrid, blk, 0, stream>>>(bufB, c_sp, src_pos, dst_pos, bufA, ER);
    k_gemm_relu<<<gemmGrid, blk, 0, stream>>>(bufA, c_dp, W1_pos, b1_pos, bufB, N);

    // 5) Layer 2, neg -> bufC
    k_zero<<<gridFor(NH, B), blk, 0, stream>>>(bufA, NH);
    k_scatter<<<scatGrid, blk, 0, stream>>>(bufC, c_sn, src_neg, dst_neg, bufA, ER);
    k_gemm_relu<<<gemmGrid, blk, 0, stream>>>(bufA, c_dn, W1_neg, b1_neg, bufC, N);

    // 6) z = z_pos - z_neg (output #1)
    k_sub<<<gridFor(NH, B), blk, 0, stream>>>(bufB, bufC, z_out, NH);

    // 7) classifier (output #2)
    k_classify<<<gridFor(EQ, B), blk, 0, stream>>>(z_out, ei, Wc, bc, cls_out, EQ);
}